// ExpEig_22643067584719
// MI455X (gfx1250) — compile-verified
//
#include <hip/hip_runtime.h>
#include <math.h>

typedef float v2f __attribute__((ext_vector_type(2)));
typedef float v8f __attribute__((ext_vector_type(8)));

#define MAT_N   128
#define LDSS    140      // padded LDS row stride (floats): conflict-free A/B/C access
#define TERMS   10       // Taylor terms with ||Y|| <= 0.5  -> fp32-accurate
#define NTHREADS 256     // 8 waves (wave32)

// D = (A * B) * invm  (+ I if addI), all 128x128 in LDS with stride LDSS.
// Wave `wave` computes rows [16*wave, 16*wave+16). 256 V_WMMA_F32_16X16X4_F32 per wave.
__device__ __forceinline__ void gemm128_lds(const float* __restrict__ Am,
                                            const float* __restrict__ Bm,
                                            float* __restrict__ Dm,
                                            float invm, int addI,
                                            int wave, int lane)
{
    const int half = lane >> 4;        // 0 or 1
    const int l15  = lane & 15;
    const int kb   = half << 1;        // this half-wave holds K = kb, kb+1

    // A fragments for the whole K=128 strip: ISA 16x4 fp32 A layout:
    //   lanes 0-15: M=lane, VGPR0=K0, VGPR1=K1 ; lanes 16-31: M=lane-16, K2,K3
    const float* arow = Am + (wave * 16 + l15) * LDSS;
    v2f afrag[32];
#pragma unroll
    for (int kc = 0; kc < 32; ++kc) {
        afrag[kc][0] = arow[kc * 4 + kb + 0];
        afrag[kc][1] = arow[kc * 4 + kb + 1];
    }

#pragma unroll
    for (int j = 0; j < 8; ++j) {
        v8f c = {};
        const float* bcol = Bm + j * 16 + l15;   // B 4x16: lane = column, halves hold K pairs
#pragma unroll
        for (int kc = 0; kc < 32; ++kc) {
            v2f b;
            b[0] = bcol[(kc * 4 + kb + 0) * LDSS];
            b[1] = bcol[(kc * 4 + kb + 1) * LDSS];
            c = __builtin_amdgcn_wmma_f32_16x16x4_f32(
                    /*neg_a=*/false, afrag[kc],
                    /*neg_b=*/false, b,
                    /*c_mod=*/(short)0, c,
                    /*reuse_a=*/false, /*reuse_b=*/false);
        }
        // C/D layout: VGPR v, lanes 0-15 -> M=v, N=lane ; lanes 16-31 -> M=v+8, N=lane-16
#pragma unroll
        for (int v = 0; v < 8; ++v) {
            const int gr = wave * 16 + v + half * 8;
            const int gc = j * 16 + l15;
            float val = c[v] * invm;
            if (addI && gr == gc) val += 1.0f;
            Dm[gr * LDSS + gc] = val;
        }
    }
}

extern "C" __global__ void __launch_bounds__(NTHREADS, 1)
expm_ss_kernel(const float* __restrict__ X, float* __restrict__ Out)
{
    extern __shared__ float lds[];
    float* Y  = lds;                      // scaled input
    float* P0 = lds + MAT_N * LDSS;      // ping
    float* P1 = lds + 2 * MAT_N * LDSS;  // pong

    __shared__ float s_red[8];
    __shared__ float s_scale;
    __shared__ int   s_sq;

    const int tid  = threadIdx.x;
    const int wave = tid >> 5;
    const int lane = tid & 31;

    const float* Xg = X   + (size_t)blockIdx.x * (MAT_N * MAT_N);
    float*       Og = Out + (size_t)blockIdx.x * (MAT_N * MAT_N);

    // ---- load X into LDS (float4), accumulate squared Frobenius norm ----
    float ss = 0.f;
    for (int i = tid; i < MAT_N * MAT_N / 4; i += NTHREADS) {
        const float4 v = ((const float4*)Xg)[i];
        const int e0 = i * 4;
        const int r  = e0 >> 7;      // /128
        const int c0 = e0 & 127;
        float* yr = Y + r * LDSS + c0;
        yr[0] = v.x; yr[1] = v.y; yr[2] = v.z; yr[3] = v.w;
        ss += v.x * v.x + v.y * v.y + v.z * v.z + v.w * v.w;
    }
    // wave32 shuffle reduction, then cross-wave via LDS
#pragma unroll
    for (int off = 16; off > 0; off >>= 1)
        ss += __shfl_down(ss, off, 32);
    if (lane == 0) s_red[wave] = ss;
    __syncthreads();
    if (tid == 0) {
        float tot = 0.f;
#pragma unroll
        for (int w = 0; w < 8; ++w) tot += s_red[w];
        float nrm = sqrtf(tot);          // ||X||_F >= ||X||_2 : conservative
        int s = 0;
        while (nrm > 0.5f && s < 48) { nrm *= 0.5f; ++s; }
        s_sq    = s;
        s_scale = ldexpf(1.0f, -s);
    }
    __syncthreads();
    const float scale = s_scale;
    const int   nsq   = s_sq;

    // ---- Y *= 2^-s ; P0 = I + Y/TERMS  (Horner seed: P_T = I folded in) ----
    for (int i = tid; i < MAT_N * MAT_N; i += NTHREADS) {
        const int r = i >> 7, c = i & 127;
        float y = Y[r * LDSS + c] * scale;
        Y[r * LDSS + c]  = y;
        P0[r * LDSS + c] = y * (1.0f / TERMS) + ((r == c) ? 1.0f : 0.0f);
    }
    __syncthreads();

    // ---- Horner: P <- I + (Y * P) / m , m = TERMS-1 .. 1 ----
    float* Pin  = P0;
    float* Pout = P1;
#pragma unroll 1
    for (int m = TERMS - 1; m >= 1; --m) {
        gemm128_lds(Y, Pin, Pout, 1.0f / (float)m, 1, wave, lane);
        __syncthreads();
        float* t = Pin; Pin = Pout; Pout = t;
    }

    // ---- nsq squarings: P <- P*P  (ping-pong on Pin/Pout; Y buffer unused) ----
    float* cur = Pin;
    float* nxt = Pout;
#pragma unroll 1
    for (int i = 0; i < nsq; ++i) {
        gemm128_lds(cur, cur, nxt, 1.0f, 0, wave, lane);
        __syncthreads();
        float* t = cur; cur = nxt; nxt = t;
    }

    // ---- store result ----
    for (int i = tid; i < MAT_N * MAT_N / 4; i += NTHREADS) {
        const int e0 = i * 4;
        const int r  = e0 >> 7;
        const int c0 = e0 & 127;
        const float* cr = cur + r * LDSS + c0;
        float4 v;
        v.x = cr[0]; v.y = cr[1]; v.z = cr[2]; v.w = cr[3];
        ((float4*)Og)[i] = v;
    }
}

extern "C" void kernel_launch(void* const* d_in, const int* in_sizes, int n_in,
                              void* d_out, int out_size, void* d_ws, size_t ws_size,
                              hipStream_t stream)
{
    const float* X  = (const float*)d_in[0];
    float*      out = (float*)d_out;
    const int batch = in_sizes[0] / (MAT_N * MAT_N);   // 4096

    const size_t smem = (size_t)3 * MAT_N * LDSS * sizeof(float);  // 210 KB < 320 KB/WGP
    (void)hipFuncSetAttribute((const void*)expm_ss_kernel,
                              hipFuncAttributeMaxDynamicSharedMemorySize, (int)smem);

    expm_ss_kernel<<<batch, NTHREADS, smem, stream>>>(X, out);
}